// KnowledgeGraphConvLayer_nBatch_geometric_1_78950088835526
// MI455X (gfx1250) — compile-verified
//
#include <hip/hip_runtime.h>
#include <hip/hip_bf16.h>

typedef float v2f __attribute__((ext_vector_type(2)));
typedef float v4f __attribute__((ext_vector_type(4)));
typedef float v8f __attribute__((ext_vector_type(8)));

#define NEG_SLOPE 0.01f
#define DCOL 128   // all GEMM outputs are 128 wide; all A halves are 128 wide

__device__ __forceinline__ float lrelu(float x) {
    return x > 0.0f ? x : NEG_SLOPE * x;
}

// -------------------------------------------------------------------------
// rootrow[i] = root_idx[batch[i]]   (tiny, done once, reused by 2 GEMMs)
// -------------------------------------------------------------------------
__global__ void k_rootrow(const int* __restrict__ batch,
                          const int* __restrict__ root_idx,
                          int* __restrict__ rootrow, int N) {
    int i = blockIdx.x * blockDim.x + threadIdx.x;
    if (i < N) rootrow[i] = root_idx[batch[i]];
}

// -------------------------------------------------------------------------
// acc[i][c] = bias[c]  (initializes scatter accumulator with the GCN bias)
// -------------------------------------------------------------------------
__global__ void k_init_bias(float* __restrict__ acc,
                            const float* __restrict__ bias, int N) {
    int i = blockIdx.x * blockDim.x + threadIdx.x;
    if (i < N * DCOL) acc[i] = bias[i & (DCOL - 1)];
}

// -------------------------------------------------------------------------
// Edge scatter: one wave per edge, each lane owns 4 channels.
// acc[dst] += w_e * xw[src]  via hardware f32 atomics (L2-resident on MI455X:
// xw + acc are 51 MB total, well under the 192 MB L2).
// -------------------------------------------------------------------------
__global__ void k_scatter(const float* __restrict__ xw,
                          const int* __restrict__ src,
                          const int* __restrict__ dst,
                          const float* __restrict__ wvals,
                          float* __restrict__ acc, int E) {
    int t = blockIdx.x * blockDim.x + threadIdx.x;
    int e = t >> 5;
    int lane = t & 31;
    if (e >= E) return;
    int s = src[e];
    int d = dst[e];
    float w = wvals[e];
    const v4f x = *(const v4f*)(xw + (size_t)s * DCOL + lane * 4);
    float* a = acc + (size_t)d * DCOL + lane * 4;
    unsafeAtomicAdd(a + 0, w * x[0]);
    unsafeAtomicAdd(a + 1, w * x[1]);
    unsafeAtomicAdd(a + 2, w * x[2]);
    unsafeAtomicAdd(a + 3, w * x[3]);
}

// -------------------------------------------------------------------------
// Fused WMMA GEMM:  C[N,128] = Avirt[N, KCHUNKS*128] @ W[KCHUNKS*128, 128]
//   chunk 0 of A comes from A0[row],           lrelu applied if LR0
//   chunk 1 of A comes from A1[rootrow[row]],  lrelu applied if LR1
//   epilogue: C = EPI ? lrelu(C + bias) : C
// Block = 256 threads = 8 waves; wave w computes rows [blk*128+w*16, +16)
// for all 128 output columns (8 v8f accumulators = 64 VGPRs).
// W staged per 128-K chunk into 64 KB LDS, interleaved (kpair,col,k&1) so a
// B fragment is one ds_load_b64 (pairs merge into ds_load_2addr_b64).
// Per k-step: 1x 8B global A load, 8 independent B fragments loaded up
// front (4x ds_load_2addr_b64 clause, single dscnt wait), then 8 WMMAs.
// -------------------------------------------------------------------------
template <int KCHUNKS, bool LR0, bool LR1, bool EPI>
__global__ void k_gemm_wmma(const float* __restrict__ A0,
                            const float* __restrict__ A1,
                            const int* __restrict__ rootrow,
                            const float* __restrict__ W,
                            const float* __restrict__ bias,
                            float* __restrict__ C, int N) {
    __shared__ float lw[128 * 128];  // 64 KB (WGP has 320 KB LDS)

    const int tid  = threadIdx.x;
    const int wave = tid >> 5;
    const int lane = tid & 31;
    const int lo   = lane & 15;   // column / row-in-halfwave index
    const int hi   = lane >> 4;   // halfwave select
    const int rowbase = blockIdx.x * 128 + wave * 16;
    const bool active = rowbase < N;  // N % 16 == 0, tiles are full or absent

    const int arow = rowbase + lo;    // A-fragment row owned by this lane
    int arow1 = 0;
    if (KCHUNKS > 1 && active) arow1 = rootrow[arow];

    v8f acc[8] = {};

#pragma unroll
    for (int kc = 0; kc < KCHUNKS; ++kc) {
        // ---- stage W chunk into LDS, interleaving adjacent K rows:
        //      lw[(k>>1)*256 + col*2 + (k&1)] = W[kc*128 + k][col]
        // Each iteration handles one (k-pair, 4-col) group: two b128 global
        // loads, four contiguous b64 LDS stores.
        const float* wsrc = W + (size_t)kc * 128 * DCOL;
        for (int i = tid; i < (128 * DCOL) / 8; i += 256) {
            int kp = i >> 5;              // k-pair 0..63
            int cg = (i & 31) * 4;        // col group base 0..124
            v4f e = *(const v4f*)(wsrc + (kp * 2 + 0) * DCOL + cg);
            v4f o = *(const v4f*)(wsrc + (kp * 2 + 1) * DCOL + cg);
            float* dstp = lw + kp * 256 + cg * 2;
#pragma unroll
            for (int j = 0; j < 4; ++j) {
                v2f p = { e[j], o[j] };
                *(v2f*)(dstp + j * 2) = p;
            }
        }
        __syncthreads();

        if (active) {
            const float* arowptr = (kc == 0) ? (A0 + (size_t)arow * DCOL)
                                             : (A1 + (size_t)arow1 * DCOL);
            const bool lr = (kc == 0) ? LR0 : LR1;
            const float* lb = lw + hi * 256 + lo * 2;  // lane's B base
#pragma unroll 4
            for (int ks = 0; ks < 32; ++ks) {
                // A frag: lane holds (row = arow, K = ks*4 + hi*2 + {0,1})
                v2f a = *(const v2f*)(arowptr + ks * 4 + hi * 2);
                if (lr) { a[0] = lrelu(a[0]); a[1] = lrelu(a[1]); }
                // B frags for all 8 column tiles, independent registers so
                // the DS loads can clause and overlap the WMMA chain.
                v2f b[8];
#pragma unroll
                for (int t = 0; t < 8; ++t)
                    b[t] = *(const v2f*)(lb + ks * 512 + t * 32);
#pragma unroll
                for (int t = 0; t < 8; ++t)
                    acc[t] = __builtin_amdgcn_wmma_f32_16x16x4_f32(
                        false, a, false, b[t], (short)0, acc[t], false, false);
            }
        }
        __syncthreads();
    }

    if (active) {
        // D layout: acc[t][r] -> row = rowbase + r + 8*hi, col = t*16 + lo
#pragma unroll
        for (int t = 0; t < 8; ++t) {
            int col = t * 16 + lo;
            float badd = EPI ? bias[col] : 0.0f;
#pragma unroll
            for (int r = 0; r < 8; ++r) {
                int row = rowbase + r + hi * 8;
                float v = acc[t][r];
                if (EPI) v = lrelu(v + badd);
                C[(size_t)row * DCOL + col] = v;
            }
        }
    }
}

// -------------------------------------------------------------------------
extern "C" void kernel_launch(void* const* d_in, const int* in_sizes, int n_in,
                              void* d_out, int out_size, void* d_ws, size_t ws_size,
                              hipStream_t stream) {
    const float* features = (const float*)d_in[0];   // [N,128]
    const float* values   = (const float*)d_in[1];   // [E]
    const float* W1       = (const float*)d_in[2];   // [128,128]
    const float* b1       = (const float*)d_in[3];   // [128]
    const float* W2       = (const float*)d_in[4];   // [256,128]
    const float* b2       = (const float*)d_in[5];   // [128]
    const float* Wl       = (const float*)d_in[6];   // [256,128]
    const float* bl       = (const float*)d_in[7];   // [128]
    const int*   adjs     = (const int*)d_in[8];     // [2,E]: row0=src,row1=dst
    const int*   root_idx = (const int*)d_in[9];     // [B]
    const int*   batch    = (const int*)d_in[12];    // [N]

    const int N = in_sizes[0] / DCOL;
    const int E = in_sizes[1];
    const int* srcI = adjs;
    const int* dstI = adjs + E;

    float* XW = (float*)d_ws;                        // [N,128] GEMM out (reused)
    float* H1 = XW + (size_t)N * DCOL;               // [N,128] conv1 accumulator
    float* H2 = H1 + (size_t)N * DCOL;               // [N,128] conv2 accumulator
    int* ROOTROW = (int*)(H2 + (size_t)N * DCOL);    // [N]

    float* OUT = (float*)d_out;                      // [N,128]

    const int gemmBlocks = (N + 127) / 128;
    const int initBlocks = (N * DCOL + 255) / 256;
    const int scatBlocks = (int)(((size_t)E * 32 + 255) / 256);

    k_rootrow<<<(N + 255) / 256, 256, 0, stream>>>(batch, root_idx, ROOTROW, N);

    // conv1: XW = features @ W1 ; H1 = b1 + scatter(values * XW[src] -> dst)
    k_gemm_wmma<1, false, false, false><<<gemmBlocks, 256, 0, stream>>>(
        features, nullptr, nullptr, W1, nullptr, XW, N);
    k_init_bias<<<initBlocks, 256, 0, stream>>>(H1, b1, N);
    k_scatter<<<scatBlocks, 256, 0, stream>>>(XW, srcI, dstI, values, H1, E);

    // conv2: A = lrelu([H1 | features[root]]) fused into GEMM A-path
    k_gemm_wmma<2, true, true, false><<<gemmBlocks, 256, 0, stream>>>(
        H1, features, ROOTROW, W2, nullptr, XW, N);
    k_init_bias<<<initBlocks, 256, 0, stream>>>(H2, b2, N);
    k_scatter<<<scatBlocks, 256, 0, stream>>>(XW, srcI, dstI, values, H2, E);

    // final linear: OUT = lrelu([lrelu(H2) | H1[root]] @ Wl + bl)
    k_gemm_wmma<2, true, false, true><<<gemmBlocks, 256, 0, stream>>>(
        H2, H1, ROOTROW, Wl, bl, OUT, N);
}